// BoundaryLoss_47545287967151
// MI455X (gfx1250) — compile-verified
//
#include <hip/hip_runtime.h>
#include <math.h>

typedef __attribute__((ext_vector_type(2))) float v2f;
typedef __attribute__((ext_vector_type(8))) float v8f;

#define Bsz 8
#define Hsz 256
#define Wsz 256
#define HW  (Hsz * Wsz)
#define NTOT (Bsz * HW)
#define BIGF 512.0f   // float(H + W)

// Workspace layout (units = 32-bit words)
#define OFF_G2IN   0                 // NTOT floats; later reused as phi
#define OFF_G2OUT  (NTOT)            // NTOT floats
#define OFF_ABSMAX (2 * NTOT)        // 8 ints (float bits of per-image max|phi|, >=0)
#define OFF_HASFG  (2 * NTOT + 8)    // 8 ints
#define OFF_PART   (2 * NTOT + 16)   // NPART floats (block partial sums)
#define NPART 512
#define RED_THREADS 256
#define ELEMS_PER_LANE (NTOT / (NPART * RED_THREADS))   // == 4, exact

// ---------------------------------------------------------------------------
// Kernel 0: zero the tiny reduction state (ws is poisoned 0xAA by the harness)
__global__ void k_init(int* absmax_bits, int* hasfg) {
    int i = threadIdx.x;
    if (i < Bsz) { absmax_bits[i] = 0; hasfg[i] = 0; }
}

// ---------------------------------------------------------------------------
// Kernel 1: per-column exact 1D vertical distance for both features.
//   feature "in"  = (target == 0)  -> dist_inside  (distance to nearest bg)
//   feature "out" = (target != 0)  -> dist_outside (distance to nearest fg)
// Backward scan stores raw distances; forward scan combines (min), clamps to
// BIG and writes the squared distance. One thread per (b, w) column.
__global__ void k_col_pass(const int* __restrict__ targets,
                           float* __restrict__ g2in, float* __restrict__ g2out,
                           int* __restrict__ hasfg) {
    int idx = blockIdx.x * blockDim.x + threadIdx.x;   // 0 .. B*W-1
    int b = idx >> 8;            // W == 256
    int w = idx & (Wsz - 1);
    int base = b * HW + w;

    // backward scan (bottom -> top)
    float din = BIGF, dout = BIGF;
    for (int h = Hsz - 1; h >= 0; --h) {
        int t = targets[base + h * Wsz];
        din  = (t == 0) ? 0.0f : din  + 1.0f;
        dout = (t != 0) ? 0.0f : dout + 1.0f;
        g2in [base + h * Wsz] = din;
        g2out[base + h * Wsz] = dout;
    }
    // forward scan (top -> bottom), combine, clamp, square
    din = BIGF; dout = BIGF;
    int anyfg = 0;
    for (int h = 0; h < Hsz; ++h) {
        int t = targets[base + h * Wsz];
        anyfg |= (t != 0);
        din  = (t == 0) ? 0.0f : din  + 1.0f;
        dout = (t != 0) ? 0.0f : dout + 1.0f;
        float gi = fminf(fminf(din,  g2in [base + h * Wsz]), BIGF);
        float go = fminf(fminf(dout, g2out[base + h * Wsz]), BIGF);
        g2in [base + h * Wsz] = gi * gi;
        g2out[base + h * Wsz] = go * go;
    }
    if (anyfg) atomicOr(&hasfg[b], 1);
}

// ---------------------------------------------------------------------------
// Kernel 2: per-row brute-force min-plus envelope (exact EDT decomposition),
// phi = sqrt(d2_out) - sqrt(d2_in); phi overwrites the g2in buffer.
// Rows are staged into LDS with CDNA5 GLOBAL_LOAD_ASYNC_TO_LDS_B32 (GVS mode:
// SGPR base + per-lane 32-bit byte offset; LDS byte address in VDST VGPR),
// tracked by ASYNCcnt -> s_wait_asynccnt, then a workgroup barrier.
// Per-image max|phi| via integer atomicMax on float bits (|phi| >= 0, exact
// and deterministic). One 256-thread block per (b, h) row.
__global__ void k_row_pass(const float* __restrict__ g2out,
                           float* __restrict__ g2in_phi,
                           int* __restrict__ absmax_bits) {
    __shared__ float sIn[Wsz];
    __shared__ float sOut[Wsz];
    __shared__ float sRed[256];

    int row = blockIdx.x;          // 0 .. B*H-1
    int b   = row >> 8;            // H == 256
    int base = row * Wsz;
    int j = threadIdx.x;

    // async global -> LDS staging (one element per lane per buffer)
    unsigned ldsIn  = (unsigned)(uintptr_t)(&sIn[j]);
    unsigned ldsOut = (unsigned)(uintptr_t)(&sOut[j]);
    unsigned goff   = (unsigned)((base + j) * 4);
    unsigned long long pIn  = (unsigned long long)(uintptr_t)g2in_phi;
    unsigned long long pOut = (unsigned long long)(uintptr_t)g2out;
    asm volatile("global_load_async_to_lds_b32 %0, %1, %2 offset:0"
                 :: "v"(ldsIn), "v"(goff), "s"(pIn) : "memory");
    asm volatile("global_load_async_to_lds_b32 %0, %1, %2 offset:0"
                 :: "v"(ldsOut), "v"(goff), "s"(pOut) : "memory");
    asm volatile("s_wait_asynccnt 0" ::: "memory");
    __syncthreads();

    float fj = (float)j;
    float dIn = 3.0e38f, dOut = 3.0e38f;
#pragma unroll 4
    for (int jp = 0; jp < Wsz; ++jp) {
        float diff = fj - (float)jp;
        float off  = diff * diff;          // exact: <= 255^2 < 2^24
        dIn  = fminf(dIn,  sIn[jp]  + off);
        dOut = fminf(dOut, sOut[jp] + off);
    }
    float phi = sqrtf(dOut) - sqrtf(dIn);
    g2in_phi[base + j] = phi;              // safe: loop read LDS copies only

    // block max of |phi|
    __syncthreads();
    sRed[j] = fabsf(phi);
    __syncthreads();
    for (int s = 128; s > 0; s >>= 1) {
        if (j < s) sRed[j] = fmaxf(sRed[j], sRed[j + s]);
        __syncthreads();
    }
    if (j == 0) atomicMax(&absmax_bits[b], __float_as_int(sRed[0]));
}

// ---------------------------------------------------------------------------
// Kernel 3: partial sums of phi_normalized * sigmoid(pred).
// Statically exact trip count (4 elems/lane) -> zero divergence anywhere, so
// EXEC is all-ones and V_WMMA_F32_16X16X4_F32 can accumulate INSIDE the loop:
//   A = {val, 0} per lane, B = all-ones, C = running tile
//   => D[M][N] += sum_K A[M][K]   (each lane's value lands in one (M,K) slot)
// After the loop, summing a lane's 8 accumulator regs covers its half's rows;
// shfl_xor(16) adds the other half -> exact f32 wave sum, independent of the
// exact A slot mapping. Per-lane add order matches a scalar accumulator.
__global__ void k_reduce1(const float* __restrict__ pred,
                          const float* __restrict__ phi,
                          const int* __restrict__ absmax_bits,
                          const int* __restrict__ hasfg,
                          float* __restrict__ partials) {
    __shared__ float sWave[8];
    int tid  = threadIdx.x;
    int gidx = blockIdx.x * RED_THREADS + tid;
    const int stride = NPART * RED_THREADS;        // 131072, compile-time

    v2f o; o.x = 1.0f; o.y = 1.0f;                 // B = all ones
    v8f c = {};
#pragma unroll
    for (int k = 0; k < ELEMS_PER_LANE; ++k) {
        int i = gidx + k * stride;                 // exact coverage of NTOT
        int b = i >> 16;                           // HW == 65536
        float denom = __int_as_float(absmax_bits[b]) + 1e-8f;
        float p = phi[i];
        p = hasfg[b] ? (p / denom) : 0.0f;
        float x = pred[i];
        float val = p * (1.0f / (1.0f + expf(-x)));
        v2f a; a.x = val; a.y = 0.0f;
        c = __builtin_amdgcn_wmma_f32_16x16x4_f32(
                /*neg_a=*/false, a, /*neg_b=*/false, o,
                /*c_mod=*/(short)0, c, /*reuse_a=*/false, /*reuse_b=*/false);
    }
    float t = c[0] + c[1] + c[2] + c[3] + c[4] + c[5] + c[6] + c[7];
    float waveSum = t + __shfl_xor(t, 16, 32);     // add other half's rows

    int lane = tid & 31;
    int wave = tid >> 5;
    if (lane == 0) sWave[wave] = waveSum;
    __syncthreads();
    if (tid == 0) {
        float s = 0.0f;
        for (int wv = 0; wv < 8; ++wv) s += sWave[wv];   // fixed order
        partials[blockIdx.x] = s;
    }
}

// ---------------------------------------------------------------------------
// Kernel 4: deterministic final tree reduction of NPART partials -> mean.
__global__ void k_reduce2(const float* __restrict__ partials, float* __restrict__ out) {
    __shared__ float s[256];
    int t = threadIdx.x;
    s[t] = partials[t] + partials[t + 256];
    __syncthreads();
    for (int k = 128; k > 0; k >>= 1) {
        if (t < k) s[t] += s[t + k];
        __syncthreads();
    }
    if (t == 0) out[0] = s[0] * (1.0f / (float)NTOT);
}

// ---------------------------------------------------------------------------
extern "C" void kernel_launch(void* const* d_in, const int* in_sizes, int n_in,
                              void* d_out, int out_size, void* d_ws, size_t ws_size,
                              hipStream_t stream) {
    const float* pred   = (const float*)d_in[0];
    const int*   target = (const int*)d_in[1];

    float* ws = (float*)d_ws;
    float* g2in   = ws + OFF_G2IN;     // reused as phi
    float* g2out  = ws + OFF_G2OUT;
    int*   absmax = (int*)(ws + OFF_ABSMAX);
    int*   hasfg  = (int*)(ws + OFF_HASFG);
    float* parts  = ws + OFF_PART;

    k_init<<<1, 64, 0, stream>>>(absmax, hasfg);
    k_col_pass<<<(Bsz * Wsz) / 256, 256, 0, stream>>>(target, g2in, g2out, hasfg);
    k_row_pass<<<Bsz * Hsz, 256, 0, stream>>>(g2out, g2in, absmax);
    k_reduce1<<<NPART, RED_THREADS, 0, stream>>>(pred, g2in, absmax, hasfg, parts);
    k_reduce2<<<1, 256, 0, stream>>>(parts, (float*)d_out);
}